// PairScorer_7997229105361
// MI455X (gfx1250) — compile-verified
//
#include <hip/hip_runtime.h>

typedef __attribute__((ext_vector_type(2))) float v2f;
typedef __attribute__((ext_vector_type(8))) float v8f;

#define NEVT 128
#define DIM  768
#define HID  150
#define HPAD 160
#define NPAIR (NEVT*(NEVT-1))   // 16256
#define LRELU_ALPHA 0.2f
#define NEGF (-9e15f)
#define EPSF 1e-6f

// ---------------------------------------------------------------- norms
__global__ void k_norm(const float* __restrict__ emb, float* __restrict__ nrm) {
    int b = blockIdx.x, i = threadIdx.x;                 // block = 128
    const float* r = emb + ((size_t)b * NEVT + i) * DIM;
    float s = 0.f;
    for (int k = 0; k < DIM; ++k) { float x = r[k]; s += x * x; }
    nrm[b * NEVT + i] = sqrtf(s);
}

// ---------------------------------------------------------------- pad weights/biases
// W2p[160][160], W3p[160][16], sb1p[160], sb2p[160]  (zero padded)
__global__ void k_prep(const float* __restrict__ sW2, const float* __restrict__ sb2,
                       const float* __restrict__ sW3, const float* __restrict__ sb1,
                       float* __restrict__ W2p, float* __restrict__ W3p,
                       float* __restrict__ sb2p, float* __restrict__ sb1p) {
    int id = blockIdx.x * 256 + threadIdx.x;
    if (id < HPAD * HPAD) {
        int r = id / HPAD, c = id - r * HPAD;
        W2p[id] = (r < HID && c < HID) ? sW2[r * HID + c] : 0.f;
    }
    int id2 = id - HPAD * HPAD;
    if (id2 >= 0 && id2 < HPAD * 16) {
        int r = id2 / 16, c = id2 - r * 16;
        W3p[id2] = (r < HID && c < 7) ? sW3[r * 7 + c] : 0.f;
    }
    int id3 = id2 - HPAD * 16;
    if (id3 >= 0 && id3 < HPAD) sb2p[id3] = (id3 < HID) ? sb2[id3] : 0.f;
    int id4 = id3 - HPAD;
    if (id4 >= 0 && id4 < HPAD) sb1p[id4] = (id4 < HID) ? sb1[id4] : 0.f;
}

// ---------------------------------------------------------------- WMMA f32 GEMM
// One wave per 16x64 C tile (4 accumulators).  C = act(A @ B[t]).
// B loads are address-clamped (never value-masked): garbage columns are
// never stored.  M mult of 16, K mult of 4.
// act: 0 none, 1 elu.  npad: if >0, store all n<npad with 0 fill for n>=N.
template <bool TRANSB>
__global__ void __launch_bounds__(32)
k_gemm(const float* __restrict__ A, long sAb, int lda,
       const float* __restrict__ B, long sBb, int ldb,
       float* __restrict__ C, long sCb, int ldc,
       int N, int K, int act, int npad) {
    int lane = threadIdx.x;                  // block = 32 (one wave)
    int n0 = blockIdx.x * 64, m0 = blockIdx.y * 16, b = blockIdx.z;
    const float* Ab = A + (size_t)b * sAb;
    const float* Bb = B + (size_t)b * sBb;
    float*       Cb = C + (size_t)b * sCb;
    int row = lane & 15, hi = lane >> 4, kb = hi * 2, col = row;

    const float* bp[4];
#pragma unroll
    for (int t = 0; t < 4; ++t) {
        int n = n0 + t * 16 + col;
        int ncl = n < N ? n : N - 1;         // address clamp only
        bp[t] = TRANSB ? (Bb + (size_t)ncl * ldb + kb)
                       : (Bb + (size_t)kb * ldb + ncl);
    }
    const long bstep = TRANSB ? 4 : 4 * (long)ldb;
    const float* ap = Ab + (size_t)(m0 + row) * lda + kb;

    v8f acc[4] = {};
    for (int k0 = 0; k0 < K; k0 += 4) {
        v2f a; a.x = ap[0]; a.y = ap[1]; ap += 4;
#pragma unroll
        for (int t = 0; t < 4; ++t) {
            v2f bv;
            if (TRANSB) { bv.x = bp[t][0]; bv.y = bp[t][1];   }
            else        { bv.x = bp[t][0]; bv.y = bp[t][ldb]; }
            acc[t] = __builtin_amdgcn_wmma_f32_16x16x4_f32(false, a, false, bv,
                                                           (short)0, acc[t], false, false);
            bp[t] += bstep;
        }
    }
#pragma unroll
    for (int t = 0; t < 4; ++t) {
        int n = n0 + t * 16 + col;
        bool ok = npad ? (n < npad) : (n < N);
        if (ok) {
#pragma unroll
            for (int e = 0; e < 8; ++e) {
                float x = acc[t][e];
                if (act == 1) x = x > 0.f ? x : expm1f(x);
                if (npad && n >= N) x = 0.f;
                Cb[(size_t)(m0 + hi * 8 + e) * ldc + n] = x;
            }
        }
    }
}

// ---------------------------------------------------------------- cosine adjacency from Gram
__global__ void k_cosadj(const float* __restrict__ gram, const float* __restrict__ nrm,
                         float* __restrict__ adjf) {
    int i = blockIdx.x, b = blockIdx.y, j = threadIdx.x;  // block = 128
    float den = fmaxf(nrm[b * NEVT + i] * nrm[b * NEVT + j], EPSF);
    float c = gram[((size_t)b * NEVT + i) * NEVT + j] / den;
    adjf[((size_t)b * NEVT + i) * NEVT + j] = (c >= 0.5f && i != j) ? 1.f : 0.f;
}

// ---------------------------------------------------------------- f1/f2 attention vectors
__global__ void k_fvec(const float* __restrict__ Wh, const float* __restrict__ a,
                       float* __restrict__ f1, float* __restrict__ f2) {
    int i = blockIdx.x, b = blockIdx.y, t = threadIdx.x;  // block = 256
    const float* r = Wh + ((size_t)b * NEVT + i) * DIM;
    __shared__ float s1[256], s2[256];
    float p1 = 0.f, p2 = 0.f;
    for (int k = t; k < DIM; k += 256) { p1 += r[k] * a[k]; p2 += r[k] * a[DIM + k]; }
    s1[t] = p1; s2[t] = p2; __syncthreads();
    for (int off = 128; off > 0; off >>= 1) {
        if (t < off) { s1[t] += s1[t + off]; s2[t] += s2[t + off]; }
        __syncthreads();
    }
    if (t == 0) { f1[b * NEVT + i] = s1[0]; f2[b * NEVT + i] = s2[0]; }
}

// ---------------------------------------------------------------- masked row softmax
__global__ void k_softmax(const float* __restrict__ f1, const float* __restrict__ f2,
                          const float* __restrict__ adjf, float* __restrict__ att) {
    int i = blockIdx.x, b = blockIdx.y, j = threadIdx.x;  // block = 128
    float e = f1[b * NEVT + i] + f2[b * NEVT + j];
    e = e > 0.f ? e : LRELU_ALPHA * e;
    float m = adjf[((size_t)b * NEVT + i) * NEVT + j];
    float val = m > 0.f ? e : NEGF;
    __shared__ float red[128];
    red[j] = val; __syncthreads();
    for (int off = 64; off > 0; off >>= 1) {
        if (j < off) red[j] = fmaxf(red[j], red[j + off]);
        __syncthreads();
    }
    float mx = red[0]; __syncthreads();
    float ex = expf(val - mx);
    red[j] = ex; __syncthreads();
    for (int off = 64; off > 0; off >>= 1) {
        if (j < off) red[j] += red[j + off];
        __syncthreads();
    }
    float sum = red[0];
    att[((size_t)b * NEVT + i) * NEVT + j] = ex / sum;
}

// ---------------------------------------------------------------- fused pair MLP (branchless)
// h1 = relu(u_i + v_j + b1p)   (16 pairs x 160, all zero-padded)
// h2 = relu(h1 @ W2p + b2p)    (WMMA, 10 N-tiles x 40 K-steps)
// out = h2 @ W3p + b3          (WMMA, cols 0..6 stored)
__global__ void __launch_bounds__(128)
k_pair(const float* __restrict__ u, const float* __restrict__ v,
       const float* __restrict__ sb1p, const float* __restrict__ W2p,
       const float* __restrict__ sb2p, const float* __restrict__ W3p,
       const float* __restrict__ sb3, float* __restrict__ out) {
    const int LD = 164;                      // LDS row stride (bank-spread)
    __shared__ float hs[4][16 * LD];
    int lane = threadIdx.x & 31;
    int w    = threadIdx.x >> 5;
    int b    = blockIdx.y;
    int tile = blockIdx.x * 4 + w;           // 1016 tiles / batch
    int p0   = tile * 16;
    const float* ub = u + (size_t)b * NEVT * HPAD;
    const float* vb = v + (size_t)b * NEVT * HPAD;
    float* h = hs[w];                        // private per-wave slice

    // stage h1 (fully unconditional: u/v/sb1p zero-padded to 160)
    for (int idx = lane; idx < 16 * HPAD; idx += 32) {
        int r = idx / HPAD, k = idx - r * HPAD;
        int p = p0 + r;
        int i = p / 127;
        int rr = p - i * 127;
        int j = rr + (rr >= i ? 1 : 0);
        h[r * LD + k] = fmaxf(ub[i * HPAD + k] + vb[j * HPAD + k] + sb1p[k], 0.f);
    }

    int row = lane & 15, hi = lane >> 4, kb = hi * 2, col = row;

    v8f acc[10] = {};
    {
        const float* pw = W2p + kb * HPAD + col;
        const float* ha = &h[row * LD + kb];
        for (int k0 = 0; k0 < HPAD; k0 += 4) {
            v2f a; a.x = ha[0]; a.y = ha[1]; ha += 4;
#pragma unroll
            for (int t = 0; t < 10; ++t) {
                v2f bv; bv.x = pw[t * 16]; bv.y = pw[t * 16 + HPAD];
                acc[t] = __builtin_amdgcn_wmma_f32_16x16x4_f32(false, a, false, bv,
                                                               (short)0, acc[t], false, false);
            }
            pw += 4 * HPAD;
        }
    }

    // relu(h2 + b2p) back into the same per-wave LDS slice (padded cols are exact zeros)
#pragma unroll
    for (int t = 0; t < 10; ++t) {
        int n = t * 16 + col;
        float bias = sb2p[n];
#pragma unroll
        for (int e = 0; e < 8; ++e) {
            int r = hi * 8 + e;
            h[r * LD + n] = fmaxf(acc[t][e] + bias, 0.f);
        }
    }

    // final layer: 160 (padded) -> 16 (7 valid)
    v8f o = {};
    {
        const float* pw3 = W3p + kb * 16 + col;
        const float* ha = &h[row * LD + kb];
        for (int k0 = 0; k0 < HPAD; k0 += 4) {
            v2f a; a.x = ha[0]; a.y = ha[1]; ha += 4;
            v2f bv; bv.x = pw3[0]; bv.y = pw3[16];
            o = __builtin_amdgcn_wmma_f32_16x16x4_f32(false, a, false, bv,
                                                      (short)0, o, false, false);
            pw3 += 64;
        }
    }
    if (col < 7) {
        float bias = sb3[col];
#pragma unroll
        for (int e = 0; e < 8; ++e) {
            int p = p0 + hi * 8 + e;
            out[((size_t)b * NPAIR + p) * 7 + col] = o[e] + bias;
        }
    }
}

// ---------------------------------------------------------------- launch
extern "C" void kernel_launch(void* const* d_in, const int* in_sizes, int n_in,
                              void* d_out, int out_size, void* d_ws, size_t ws_size,
                              hipStream_t stream) {
    (void)in_sizes; (void)n_in; (void)out_size; (void)ws_size;
    const float* emb = (const float*)d_in[0];   // [8,128,768]
    const float* W1  = (const float*)d_in[1];   // [768,768]
    const float* a1  = (const float*)d_in[2];   // [1536,1]
    const float* W2  = (const float*)d_in[3];
    const float* a2  = (const float*)d_in[4];
    const float* sW1 = (const float*)d_in[5];   // [1536,150]
    const float* sb1 = (const float*)d_in[6];
    const float* sW2 = (const float*)d_in[7];   // [150,150]
    const float* sb2 = (const float*)d_in[8];
    const float* sW3 = (const float*)d_in[9];   // [150,7]
    const float* sb3 = (const float*)d_in[10];
    float* out = (float*)d_out;

    float* ws   = (float*)d_ws;
    float* nrm  = ws;                 //   1024
    float* adjf = nrm  + 1024;        // 131072
    float* f1   = adjf + 131072;      //   1024
    float* f2   = f1   + 1024;        //   1024
    float* att  = f2   + 1024;        // 131072 (also holds Gram matrix)
    float* Wh   = att  + 131072;      // 786432
    float* x    = Wh   + 786432;      // 786432
    float* uu   = x    + 786432;      // 163840 (ld = 160, zero-padded)
    float* vv   = uu   + 163840;      // 163840
    float* W2p  = vv   + 163840;      //  25600
    float* W3p  = W2p  + 25600;       //   2560
    float* sb2p = W3p  + 2560;        //    160
    float* sb1p = sb2p + 160;         //    160

    const long sE = (long)NEVT * DIM;       // 98304
    const long sA = (long)NEVT * NEVT;      // 16384
    const long sU = (long)NEVT * HPAD;      // 20480

    k_prep<<<112, 256, 0, stream>>>(sW2, sb2, sW3, sb1, W2p, W3p, sb2p, sb1p);
    k_norm<<<8, 128, 0, stream>>>(emb, nrm);

    // Gram = emb @ emb^T  (WMMA, transposed B)
    k_gemm<true><<<dim3(2, 8, 8), 32, 0, stream>>>(emb, sE, DIM, emb, sE, DIM,
                                                   att, sA, NEVT, NEVT, DIM, 0, 0);
    k_cosadj<<<dim3(NEVT, 8), 128, 0, stream>>>(att, nrm, adjf);

    // Wh1 = emb @ W1
    k_gemm<false><<<dim3(12, 8, 8), 32, 0, stream>>>(emb, sE, DIM, W1, 0, DIM,
                                                     Wh, sE, DIM, DIM, DIM, 0, 0);
    k_fvec<<<dim3(NEVT, 8), 256, 0, stream>>>(Wh, a1, f1, f2);
    k_softmax<<<dim3(NEVT, 8), 128, 0, stream>>>(f1, f2, adjf, att);
    // x1 = elu(att @ Wh1)
    k_gemm<false><<<dim3(12, 8, 8), 32, 0, stream>>>(att, sA, NEVT, Wh, sE, DIM,
                                                     x, sE, DIM, DIM, NEVT, 1, 0);
    // Wh2 = x1 @ W2
    k_gemm<false><<<dim3(12, 8, 8), 32, 0, stream>>>(x, sE, DIM, W2, 0, DIM,
                                                     Wh, sE, DIM, DIM, DIM, 0, 0);
    k_fvec<<<dim3(NEVT, 8), 256, 0, stream>>>(Wh, a2, f1, f2);
    k_softmax<<<dim3(NEVT, 8), 128, 0, stream>>>(f1, f2, adjf, att);
    // x2 = att @ Wh2
    k_gemm<false><<<dim3(12, 8, 8), 32, 0, stream>>>(att, sA, NEVT, Wh, sE, DIM,
                                                     x, sE, DIM, DIM, NEVT, 0, 0);
    // u = x2 @ sW1[:768], v = x2 @ sW1[768:]  (stored ld=160, zero-padded)
    k_gemm<false><<<dim3(3, 8, 8), 32, 0, stream>>>(x, sE, DIM, sW1, 0, HID,
                                                    uu, sU, HPAD, HID, DIM, 0, HPAD);
    k_gemm<false><<<dim3(3, 8, 8), 32, 0, stream>>>(x, sE, DIM, sW1 + (size_t)DIM * HID, 0, HID,
                                                    vv, sU, HPAD, HID, DIM, 0, HPAD);
    // fused pair scorer
    k_pair<<<dim3(254, 8), 128, 0, stream>>>(uu, vv, sb1p, W2p, sb2p, W3p, sb3, out);
}